// MetricLoss_48404281426340
// MI455X (gfx1250) — compile-verified
//
#include <hip/hip_runtime.h>
#include <hip/hip_bf16.h>

typedef __attribute__((ext_vector_type(2))) float v2f;
typedef __attribute__((ext_vector_type(8))) float v8f;

// Problem shape from reference setup_inputs(): fixed by the harness.
constexpr int B_   = 8;
constexpr int N_   = 512 * 512;   // H*W pixels per image
constexpr int D_   = 32;          // embedding dim
constexpr int K_   = 64;          // instances
constexpr float MARGIN = 0.25f;

constexpr int TPB = 256;                         // 8 waves (wave32)
constexpr int PIX_PER_BLOCK = 4096;              // 16 pixels / thread
constexpr int BLOCKS_PER_IMG = N_ / PIX_PER_BLOCK; // 64

// Workspace layout (floats):
//   gsum : [B][D][K]  (d-major so LDS bank = k, conflict-free across random labels)
//   gcnt : [B][K]
constexpr int GSUM_ELEMS = B_ * D_ * K_;
constexpr int GCNT_ELEMS = B_ * K_;
constexpr int WS_ELEMS   = GSUM_ELEMS + GCNT_ELEMS;

__global__ void zero_ws_kernel(float* ws, int n) {
    int i = blockIdx.x * blockDim.x + threadIdx.x;
    if (i < n) ws[i] = 0.0f;
}

// Stage 1: bandwidth-bound segment reduction.
// Coalesced b128 streams from HBM; LDS-atomic binning acc[d][k]; one global
// atomic flush per block. ~264 MB traffic -> ~11.3 us floor at 23.3 TB/s.
__global__ void __launch_bounds__(TPB)
seg_reduce_kernel(const float* __restrict__ emb, const int* __restrict__ lab,
                  float* __restrict__ gsum, float* __restrict__ gcnt) {
    __shared__ float acc[D_ * K_]; // [d][k] : bank index = k (random) -> low conflict
    __shared__ float cnt[K_];

    const int tid = threadIdx.x;
    for (int i = tid; i < D_ * K_; i += TPB) acc[i] = 0.0f;
    if (tid < K_) cnt[tid] = 0.0f;
    __syncthreads();

    const int b     = blockIdx.x / BLOCKS_PER_IMG;
    const int chunk = blockIdx.x % BLOCKS_PER_IMG;
    const float4* e4 = reinterpret_cast<const float4*>(emb + (size_t)b * N_ * D_);
    const int*    lb = lab + (size_t)b * N_;
    const int base = chunk * PIX_PER_BLOCK;

    for (int it = 0; it < PIX_PER_BLOCK / TPB; ++it) {
        const int p = base + it * TPB + tid;
        const int l = lb[p];
#pragma unroll
        for (int j = 0; j < D_ / 4; ++j) {          // 8x global_load_b128 per pixel
            float4 v = e4[(size_t)p * (D_ / 4) + j];
            const int d0 = j * 4;
            atomicAdd(&acc[(d0 + 0) * K_ + l], v.x); // ds_add_f32 (no return)
            atomicAdd(&acc[(d0 + 1) * K_ + l], v.y);
            atomicAdd(&acc[(d0 + 2) * K_ + l], v.z);
            atomicAdd(&acc[(d0 + 3) * K_ + l], v.w);
        }
        atomicAdd(&cnt[l], 1.0f);
    }
    __syncthreads();

    float* gs = gsum + (size_t)b * D_ * K_;
    for (int i = tid; i < D_ * K_; i += TPB) atomicAdd(&gs[i], acc[i]);
    if (tid < K_) atomicAdd(&gcnt[b * K_ + tid], cnt[tid]);
}

// Stage 2: centroids -> masked hinge^2 matrix S (64x64, in LDS) per batch,
// then reduce Sum(S) on the matrix pipe with exact-f32 WMMA 16x16x4:
// accumulate ones(16x4) x S_tile(4x16) over all 16 row-chunks x 4 col-blocks
// into one C; every row of C then holds the grand column-sums, so
// total = sum_n C[0][n].
__global__ void __launch_bounds__(TPB)
finalize_kernel(const float* __restrict__ gsum, const float* __restrict__ gcnt,
                float* __restrict__ out) {
    __shared__ float cent[K_ * D_];  // [k][d]
    __shared__ float S[K_ * K_];
    __shared__ float pres[K_];
    __shared__ float red[16];
    __shared__ float batch_acc;

    const int tid = threadIdx.x;
    if (tid == 0) batch_acc = 0.0f;

    for (int b = 0; b < B_; ++b) {
        __syncthreads();
        // presence + centroids
        for (int i = tid; i < K_; i += TPB)
            pres[i] = (gcnt[b * K_ + i] > 0.5f) ? 1.0f : 0.0f;
        for (int i = tid; i < K_ * D_; i += TPB) {
            const int k = i / D_, d = i % D_;
            const float c = gcnt[b * K_ + k];
            cent[i] = gsum[(size_t)b * D_ * K_ + d * K_ + k] / fmaxf(c, 1.0f);
        }
        __syncthreads();

        // masked squared-hinge pair matrix (strict upper triangle, both present)
        for (int i = tid; i < K_ * K_; i += TPB) {
            const int r = i / K_, c = i % K_;
            float s = 0.0f;
            if (r < c && pres[r] > 0.0f && pres[c] > 0.0f) {
                float dsum = 0.0f;
#pragma unroll
                for (int d = 0; d < D_; ++d)
                    dsum += fabsf(cent[r * D_ + d] - cent[c * D_ + d]);
                const float h = fmaxf(MARGIN - dsum, 0.0f);
                s = h * h;
            }
            S[i] = s;
        }
        __syncthreads();

        // WMMA reduction by wave 0 (EXEC all-ones: whole wave enters).
        if (tid < 32) {
            const int lane = tid;
            const int n    = lane & 15;
            const int half = lane >> 4;          // A: 16x4 ones -> lanes 0..15 M, VGPRv=K
            v2f a; a.x = 1.0f; a.y = 1.0f;
            v8f c = {};
            for (int rb = 0; rb < 16; ++rb) {    // 4-row chunks
                for (int cb = 0; cb < 4; ++cb) { // 16-col blocks
                    v2f bv;                       // B: 4x16, VGPRv row k = 2*v + half
                    bv.x = S[(rb * 4 + 0 + half * 1) * K_ + cb * 16 + n];
                    bv.y = S[(rb * 4 + 2 + half * 1) * K_ + cb * 16 + n];
                    c = __builtin_amdgcn_wmma_f32_16x16x4_f32(
                            false, a, false, bv, (short)0, c, false, false);
                }
            }
            if (lane < 16) red[lane] = c[0];     // C row M=0: VGPR0, lanes 0..15
        }
        __syncthreads();

        if (tid == 0) {
            float tot = 0.0f;
            for (int i = 0; i < 16; ++i) tot += red[i];
            float np = 0.0f;
            for (int k = 0; k < K_; ++k) np += pres[k];
            const float ncomp = np * (np - 1.0f) * 0.5f;
            batch_acc += tot / ncomp;
        }
    }
    __syncthreads();
    if (tid == 0) out[0] = batch_acc / (float)B_;
}

extern "C" void kernel_launch(void* const* d_in, const int* in_sizes, int n_in,
                              void* d_out, int out_size, void* d_ws, size_t ws_size,
                              hipStream_t stream) {
    (void)in_sizes; (void)n_in; (void)out_size; (void)ws_size;
    const float* emb = (const float*)d_in[0];
    const int*   lab = (const int*)d_in[1];
    float* gsum = (float*)d_ws;
    float* gcnt = gsum + GSUM_ELEMS;
    float* out  = (float*)d_out;

    zero_ws_kernel<<<(WS_ELEMS + TPB - 1) / TPB, TPB, 0, stream>>>((float*)d_ws, WS_ELEMS);
    seg_reduce_kernel<<<B_ * BLOCKS_PER_IMG, TPB, 0, stream>>>(emb, lab, gsum, gcnt);
    finalize_kernel<<<1, TPB, 0, stream>>>(gsum, gcnt, out);
}